// BondPredictor_12094627906354
// MI455X (gfx1250) — compile-verified
//
#include <hip/hip_runtime.h>
#include <math.h>

#define NG    288
#define NLIG  32
#define NPRO  256
#define KNN   24
#define BATCH 16
#define NNODE 4608           // BATCH*NG
#define EKNN  110592         // NNODE*KNN
#define HPER  496
#define HH    7936           // BATCH*HPER
#define ELIG  15872          // 2*HH
#define ETOT  126464         // EKNN + ELIG
#define NDIM  256
#define EDIM  128

typedef float v2f __attribute__((ext_vector_type(2)));
typedef float v8f __attribute__((ext_vector_type(8)));

// ---------------------------------------------------------------------------
// WMMA helpers: fp32 16x16x4 tiles.
// A (16x4): lane = row (l&15), VGPR pair holds K = {2*(l>>4), 2*(l>>4)+1}
// B (4x16): lane = col (l&15), VGPR pair holds K rows {2*(l>>4), 2*(l>>4)+1}
// C/D (16x16): vgpr v, lane l -> row = v + 8*(l>>4), col = l&15
//
// Weights are pre-packed as Wp[pair][col] = {W[2p][col], W[2p+1][col]}
// (zero padded to K multiple of 4) so each B fragment is one b64 load.
// ---------------------------------------------------------------------------
__device__ __forceinline__ v8f wmma4(v2f a, v2f b, v8f c) {
  return __builtin_amdgcn_wmma_f32_16x16x4_f32(false, a, false, b, (short)0, c,
                                               false, false);
}

// one 16x16 output tile
__device__ __forceinline__ v8f gemm_one(const float* As, int lda,
                                        const float2* __restrict__ Wp, int N,
                                        int Kpad4, int cb, v8f acc) {
  const int lane = threadIdx.x & 31;
  const int sub  = lane & 15;
  const int kh   = lane >> 4;
  const float* arow = As + sub * lda;
  const float2* wc  = Wp + cb + sub;
  for (int k = 0; k < Kpad4; k += 4) {
    const int p = (k >> 1) + kh;
    v2f a; a.x = arow[2 * p]; a.y = arow[2 * p + 1];
    const float2 bw = wc[(size_t)p * N];
    v2f b; b.x = bw.x; b.y = bw.y;
    acc = wmma4(a, b, acc);
  }
  return acc;
}

// two 16x16 output tiles (cols cb0, cb1) sharing A fragments
__device__ __forceinline__ void gemm_pair(const float* As, int lda,
                                          const float2* __restrict__ Wp, int N,
                                          int Kpad4, int cb0, int cb1,
                                          v8f& acc0, v8f& acc1) {
  const int lane = threadIdx.x & 31;
  const int sub  = lane & 15;
  const int kh   = lane >> 4;
  const float* arow = As + sub * lda;
  const float2* w0  = Wp + cb0 + sub;
  const float2* w1  = Wp + cb1 + sub;
  for (int k = 0; k < Kpad4; k += 4) {
    const int p = (k >> 1) + kh;
    v2f a; a.x = arow[2 * p]; a.y = arow[2 * p + 1];
    const float2 bw0 = w0[(size_t)p * N];
    const float2 bw1 = w1[(size_t)p * N];
    v2f b0; b0.x = bw0.x; b0.y = bw0.y;
    v2f b1; b1.x = bw1.x; b1.y = bw1.y;
    acc0 = wmma4(a, b0, acc0);
    acc1 = wmma4(a, b1, acc1);
  }
}

__device__ __forceinline__ float silu_f(float x) {
  return x / (1.0f + expf(-x));
}

// store C tile into LDS with bias + activation (0=none,1=silu,2=relu)
__device__ __forceinline__ void frag_store_act(float* Hs, int ldh, int colbase,
                                               v8f c, const float* __restrict__ bias,
                                               int act) {
  const int lane = threadIdx.x & 31;
  const int sub  = lane & 15;
  const int rb   = (lane >> 4) * 8;
  const float bv = bias[colbase + sub];
#pragma unroll
  for (int v = 0; v < 8; v++) {
    float x = c[v] + bv;
    if (act == 1) x = silu_f(x);
    else if (act == 2) x = fmaxf(x, 0.0f);
    Hs[(rb + v) * ldh + colbase + sub] = x;
  }
}

// store C tile to global rows r0.. with bias
__device__ __forceinline__ void frag_store_global(float* __restrict__ out, int ldo,
                                                  int colbase, v8f c,
                                                  const float* __restrict__ bias) {
  const int lane = threadIdx.x & 31;
  const int sub  = lane & 15;
  const int rb   = (lane >> 4) * 8;
  const float bv = bias[colbase + sub];
#pragma unroll
  for (int v = 0; v < 8; v++)
    out[(size_t)(rb + v) * ldo + colbase + sub] = c[v] + bv;
}

__device__ __forceinline__ float gauss_time(float tt, int k) {
  const float off = (float)k * (1000.0f / 15.0f);
  const float d = tt - off;
  return expf(-1.125e-4f * d * d);   // -0.5/(1000/15)^2 exactly
}

__device__ __forceinline__ int triu_off(int i) { return (i * (63 - i)) / 2; }

// ---------------------------------------------------------------------------
// 0) weight packing: Wp[p][c] = {W[2p][c], W[2p+1][c]}, zero padded
// ---------------------------------------------------------------------------
__global__ void k_pack_w(const float* __restrict__ W, float2* __restrict__ Wp,
                         int Kin, int N, int npairs) {
  const int idx = blockIdx.x * 256 + threadIdx.x;
  if (idx >= npairs * N) return;
  const int p = idx / N, c = idx % N;
  const int k0 = 2 * p, k1 = 2 * p + 1;
  const float x = (k0 < Kin) ? W[(size_t)k0 * N + c] : 0.0f;
  const float y = (k1 < Kin) ? W[(size_t)k1 * N + c] : 0.0f;
  Wp[idx] = make_float2(x, y);
}

// ---------------------------------------------------------------------------
// 1) node embedding: h[NNODE x 256], pos[NNODE x 4]
// ---------------------------------------------------------------------------
__global__ void k_embed_nodes(const float* __restrict__ pnode,
                              const float* __restrict__ ppos,
                              const float* __restrict__ lnode,
                              const float* __restrict__ lpos,
                              const int* __restrict__ t,
                              const float* __restrict__ Wpn,   // 27x255
                              const float* __restrict__ Wln,   // 13x239
                              float* __restrict__ h, float* __restrict__ pos) {
  const int n = blockIdx.x;
  const int b = n / NG, r = n % NG;
  const int d = threadIdx.x;
  float v = 0.0f;
  if (r < NLIG) {
    const int li = b * NLIG + r;
    if (d < 239) {
      float s = 0.0f;
      for (int k = 0; k < 13; k++) s += lnode[li * 13 + k] * Wln[k * 239 + d];
      v = s;
    } else if (d < 255) {
      v = gauss_time((float)t[b], d - 239);
    } else {
      v = 1.0f;
    }
    if (d < 3) pos[n * 4 + d] = lpos[li * 3 + d];
    if (d == 3) pos[n * 4 + 3] = 0.0f;
  } else {
    const int pi = b * NPRO + (r - NLIG);
    if (d < 255) {
      float s = 0.0f;
      for (int k = 0; k < 27; k++) s += pnode[pi * 27 + k] * Wpn[k * 255 + d];
      v = s;
    } else {
      v = 0.0f;
    }
    if (d < 3) pos[n * 4 + d] = ppos[pi * 3 + d];
    if (d == 3) pos[n * 4 + 3] = 0.0f;
  }
  h[(size_t)n * NDIM + d] = v;
}

// ---------------------------------------------------------------------------
// 2) KNN: per node, 24 nearest (ascending d2, tie -> lowest index)
// ---------------------------------------------------------------------------
__global__ void k_knn(const float* __restrict__ pos, const int* __restrict__ t,
                      int* __restrict__ src, int* __restrict__ dst,
                      float* __restrict__ vm, float* __restrict__ et) {
  __shared__ float px[NG], py[NG], pz[NG];
  __shared__ float d2s[NG];
  __shared__ float rv[NG];
  __shared__ int   ri[NG];
  __shared__ int   nbr[KNN];
  const int n = blockIdx.x;
  const int b = n / NG, r = n % NG;
  const int tid = threadIdx.x;
  {
    const int g = b * NG + tid;
    px[tid] = pos[g * 4 + 0];
    py[tid] = pos[g * 4 + 1];
    pz[tid] = pos[g * 4 + 2];
  }
  __syncthreads();
  {
    const float dx = px[tid] - px[r];
    const float dy = py[tid] - py[r];
    const float dz = pz[tid] - pz[r];
    const float dd = dx * dx + dy * dy + dz * dz;
    d2s[tid] = (tid == r) ? 1.0e10f : dd;
  }
  __syncthreads();
  for (int k = 0; k < KNN; k++) {
    rv[tid] = d2s[tid];
    ri[tid] = tid;
    __syncthreads();
    if (tid < 32) {  // fold 256..287 into 0..31
      const float v2 = rv[tid + 256]; const int i2 = ri[tid + 256];
      if (v2 < rv[tid] || (v2 == rv[tid] && i2 < ri[tid])) { rv[tid] = v2; ri[tid] = i2; }
    }
    __syncthreads();
    for (int s = 128; s >= 1; s >>= 1) {
      if (tid < s) {
        const float v2 = rv[tid + s]; const int i2 = ri[tid + s];
        if (v2 < rv[tid] || (v2 == rv[tid] && i2 < ri[tid])) { rv[tid] = v2; ri[tid] = i2; }
      }
      __syncthreads();
    }
    if (tid == 0) { nbr[k] = ri[0]; d2s[ri[0]] = 3.0e38f; }
    __syncthreads();
  }
  if (tid < KNN) {
    const int eid = n * KNN + tid;
    const int nb = nbr[tid];
    src[eid] = n;
    dst[eid] = b * NG + nb;
    vm[eid]  = ((r < NLIG) && (nb < NLIG)) ? 0.0f : 1.0f;
    et[eid]  = (float)t[b] * 0.001f;
  }
}

// 3) init e rows for KNN edges: e = vm * W_pe[0]
__global__ void k_init_knn_edges(const float* __restrict__ vm,
                                 const float* __restrict__ Wpe0,
                                 float* __restrict__ e) {
  const size_t idx = (size_t)blockIdx.x * 256 + threadIdx.x;  // EKNN*128 total
  const int row = (int)(idx >> 7);
  const int col = (int)(idx & 127);
  e[idx] = vm[row] * Wpe0[col];
}

// 4) ligand edge embedding + indices
__global__ void k_embed_ligand_edges(const float* __restrict__ lnode,
                                     const int* __restrict__ lei,   // 2 x 15872
                                     const int* __restrict__ leb,   // 15872
                                     const int* __restrict__ t,
                                     const float* __restrict__ Wle, // 26x112
                                     float* __restrict__ e,
                                     int* __restrict__ src, int* __restrict__ dst,
                                     float* __restrict__ vm, float* __restrict__ et) {
  const int j = blockIdx.x;
  const int c = threadIdx.x;
  const int ls = lei[j];
  const int ld = lei[ELIG + j];
  float v;
  if (c < 112) {
    float s = 0.0f;
    for (int k = 0; k < 13; k++) s += lnode[ls * 13 + k] * Wle[k * 112 + c];
    for (int k = 0; k < 13; k++) s += lnode[ld * 13 + k] * Wle[(13 + k) * 112 + c];
    v = s;
  } else {
    v = gauss_time((float)t[leb[j]], c - 112);
  }
  e[(size_t)(EKNN + j) * EDIM + c] = v;
  if (c == 0) {
    const int eid = EKNN + j;
    src[eid] = (ls / NLIG) * NG + (ls % NLIG);
    dst[eid] = (ld / NLIG) * NG + (ld % NLIG);
    vm[eid]  = 1.0f;
    et[eid]  = (float)t[leb[j]] * 0.001f;
  }
}

// 5) per-layer: edge geometry (dist2)
__global__ void k_edge_geom(const float* __restrict__ pos,
                            const int* __restrict__ src, const int* __restrict__ dst,
                            float* __restrict__ dist2) {
  const int idx = blockIdx.x * 256 + threadIdx.x;
  if (idx >= ETOT) return;
  const int s = src[idx], d = dst[idx];
  const float rx = pos[s * 4 + 0] - pos[d * 4 + 0];
  const float ry = pos[s * 4 + 1] - pos[d * 4 + 1];
  const float rz = pos[s * 4 + 2] - pos[d * 4 + 2];
  dist2[idx] = rx * rx + ry * ry + rz * rz;
}

// 6) message MLP: m = silu([h_src,h_dst,e,d2,et] @ Wm1 + bm1) @ Wm2 + bm2
__global__ void __launch_bounds__(256)
k_msg(const float* __restrict__ h, const float* __restrict__ e,
      const float* __restrict__ dist2, const float* __restrict__ et,
      const int* __restrict__ src, const int* __restrict__ dst,
      const float2* __restrict__ Wm1p, const float* __restrict__ bm1,
      const float2* __restrict__ Wm2p, const float* __restrict__ bm2,
      float* __restrict__ m) {
  constexpr int LDA = 646, LDH = 260;   // Kin=642 -> Kpad4=644
  __shared__ float As[16 * LDA];
  __shared__ float Hs[16 * LDH];
  __shared__ int sidx[16], didx[16];
  const int tid = threadIdx.x;
  const int r0 = blockIdx.x * 16;
  if (tid < 16) { sidx[tid] = src[r0 + tid]; didx[tid] = dst[r0 + tid]; }
  __syncthreads();
  for (int idx = tid; idx < 16 * LDA; idx += 256) {
    const int row = idx / LDA, col = idx % LDA;
    const int er = r0 + row;
    float v;
    if (col < 256)        v = h[(size_t)sidx[row] * NDIM + col];
    else if (col < 512)   v = h[(size_t)didx[row] * NDIM + (col - 256)];
    else if (col < 640)   v = e[(size_t)er * EDIM + (col - 512)];
    else if (col == 640)  v = dist2[er];
    else if (col == 641)  v = et[er];
    else                  v = 0.0f;
    As[idx] = v;
  }
  __syncthreads();
  const int w = tid >> 5;
  v8f c0 = {}, c1 = {};
  gemm_pair(As, LDA, Wm1p, 256, 644, w * 16, 128 + w * 16, c0, c1);
  frag_store_act(Hs, LDH, w * 16, c0, bm1, 1);
  frag_store_act(Hs, LDH, 128 + w * 16, c1, bm1, 1);
  __syncthreads();
  v8f d0 = {}, d1 = {};
  gemm_pair(Hs, LDH, Wm2p, 256, 256, w * 16, 128 + w * 16, d0, d1);
  frag_store_global(m + (size_t)r0 * 256, 256, w * 16, d0, bm2);
  frag_store_global(m + (size_t)r0 * 256, 256, 128 + w * 16, d1, bm2);
}

// 7) edge MLP (in place): e = silu([e,m] @ We1 + be1) @ We2 + be2
__global__ void __launch_bounds__(256)
k_eupd(float* __restrict__ e, const float* __restrict__ m,
       const float2* __restrict__ We1p, const float* __restrict__ be1,
       const float2* __restrict__ We2p, const float* __restrict__ be2) {
  constexpr int LDA = 390, LDH = 132;   // Kin=384
  __shared__ float As[16 * LDA];
  __shared__ float Hs[16 * LDH];
  const int tid = threadIdx.x;
  const int r0 = blockIdx.x * 16;
  for (int idx = tid; idx < 16 * LDA; idx += 256) {
    const int row = idx / LDA, col = idx % LDA;
    const int er = r0 + row;
    float v;
    if (col < 128)      v = e[(size_t)er * EDIM + col];
    else if (col < 384) v = m[(size_t)er * 256 + (col - 128)];
    else                v = 0.0f;
    As[idx] = v;
  }
  __syncthreads();
  const int w = tid >> 5;
  v8f c0 = {};
  c0 = gemm_one(As, LDA, We1p, 128, 384, w * 16, c0);
  frag_store_act(Hs, LDH, w * 16, c0, be1, 1);
  __syncthreads();
  v8f d0 = {};
  d0 = gemm_one(Hs, LDH, We2p, 128, 128, w * 16, d0);
  frag_store_global(e + (size_t)r0 * EDIM, EDIM, w * 16, d0, be2);
}

// 8) deterministic aggregation: agg[n] = sum_{edges with src=n} m*vm
__global__ void k_agg(const float* __restrict__ m, const float* __restrict__ vm,
                      float* __restrict__ agg) {
  const int n = blockIdx.x;
  const int d = threadIdx.x;
  const int b = n / NG, r = n % NG;
  float s = 0.0f;
  for (int k = 0; k < KNN; k++) {
    const int eid = n * KNN + k;
    s += m[(size_t)eid * 256 + d] * vm[eid];
  }
  if (r < NLIG) {
    const int i = r;
    for (int j = i + 1; j < NLIG; j++) {
      const int eid = EKNN + b * HPER + triu_off(i) + (j - i - 1);
      s += m[(size_t)eid * 256 + d];
    }
    for (int j = 0; j < i; j++) {
      const int eid = EKNN + HH + b * HPER + triu_off(j) + (i - j - 1);
      s += m[(size_t)eid * 256 + d];
    }
  }
  agg[(size_t)n * 256 + d] = s;
}

// 9) node MLP with residual: h += silu([h,agg,nt] @ Wh1 + bh1) @ Wh2 + bh2
__global__ void __launch_bounds__(256)
k_hupd(float* __restrict__ h, const float* __restrict__ agg,
       const int* __restrict__ t,
       const float2* __restrict__ Wh1p, const float* __restrict__ bh1,
       const float2* __restrict__ Wh2p, const float* __restrict__ bh2) {
  constexpr int LDA = 518, LDH = 260;   // Kin=513 -> Kpad4=516
  __shared__ float As[16 * LDA];
  __shared__ float Hs[16 * LDH];
  const int tid = threadIdx.x;
  const int r0 = blockIdx.x * 16;
  for (int idx = tid; idx < 16 * LDA; idx += 256) {
    const int row = idx / LDA, col = idx % LDA;
    const int n = r0 + row;
    float v;
    if (col < 256)       v = h[(size_t)n * NDIM + col];
    else if (col < 512)  v = agg[(size_t)n * 256 + (col - 256)];
    else if (col == 512) v = (float)t[n / NG] * 0.001f;
    else                 v = 0.0f;
    As[idx] = v;
  }
  __syncthreads();
  const int w = tid >> 5;
  v8f c0 = {}, c1 = {};
  gemm_pair(As, LDA, Wh1p, 256, 516, w * 16, 128 + w * 16, c0, c1);
  frag_store_act(Hs, LDH, w * 16, c0, bh1, 1);
  frag_store_act(Hs, LDH, 128 + w * 16, c1, bh1, 1);
  __syncthreads();
  v8f d0 = {}, d1 = {};
  gemm_pair(Hs, LDH, Wh2p, 256, 256, w * 16, 128 + w * 16, d0, d1);
  const int lane = tid & 31, sub = lane & 15, rb = (lane >> 4) * 8;
  const float b0 = bh2[w * 16 + sub];
  const float b1 = bh2[128 + w * 16 + sub];
#pragma unroll
  for (int v = 0; v < 8; v++) {
    const size_t row = (size_t)(r0 + rb + v);
    h[row * NDIM + w * 16 + sub]       += d0[v] + b0;
    h[row * NDIM + 128 + w * 16 + sub] += d1[v] + b1;
  }
}

// 10) coef = (m @ Wx) * vm   (one wave per edge)
__global__ void k_coef(const float* __restrict__ m, const float* __restrict__ Wx,
                       const float* __restrict__ vm, float* __restrict__ coef) {
  const int e = blockIdx.x * 8 + (threadIdx.x >> 5);
  const int lane = threadIdx.x & 31;
  float s = 0.0f;
  for (int j = 0; j < 8; j++)
    s += m[(size_t)e * 256 + lane + 32 * j] * Wx[lane + 32 * j];
  for (int off = 16; off; off >>= 1) s += __shfl_down(s, off, 32);
  if (lane == 0) coef[e] = s * vm[e];
}

// 11) position update (deterministic per-node gather)
__global__ void k_pos(const float* __restrict__ posA, float* __restrict__ posB,
                      const float* __restrict__ dist2, const float* __restrict__ coef,
                      const int* __restrict__ dst) {
  const int n = blockIdx.x * 256 + threadIdx.x;
  if (n >= NNODE) return;
  const int b = n / NG, r = n % NG;
  const float sx = posA[n * 4 + 0], sy = posA[n * 4 + 1], sz = posA[n * 4 + 2];
  float ux = 0.0f, uy = 0.0f, uz = 0.0f;
#define ACC(eid)                                                         \
  {                                                                      \
    const int _e = (eid);                                                \
    const int dn = dst[_e];                                              \
    const float c = coef[_e] / (sqrtf(dist2[_e]) + 1.0f);                \
    ux += (sx - posA[dn * 4 + 0]) * c;                                   \
    uy += (sy - posA[dn * 4 + 1]) * c;                                   \
    uz += (sz - posA[dn * 4 + 2]) * c;                                   \
  }
  for (int k = 0; k < KNN; k++) ACC(n * KNN + k);
  if (r < NLIG) {
    const int i = r;
    for (int j = i + 1; j < NLIG; j++) ACC(EKNN + b * HPER + triu_off(i) + (j - i - 1));
    for (int j = 0; j < i; j++)        ACC(EKNN + HH + b * HPER + triu_off(j) + (i - j - 1));
  }
#undef ACC
  const float lm = (r < NLIG) ? 1.0f : 0.0f;
  posB[n * 4 + 0] = sx + ux * lm;
  posB[n * 4 + 1] = sy + uy * lm;
  posB[n * 4 + 2] = sz + uz * lm;
  posB[n * 4 + 3] = 0.0f;
}

// 12) decoder: relu(ext@W1+b1) -> relu(@W2+b2) -> @W3+b3
__global__ void __launch_bounds__(256)
k_decoder(const float* __restrict__ h, const float* __restrict__ e,
          const int* __restrict__ lei,
          const float2* __restrict__ W1p, const float* __restrict__ b1,
          const float2* __restrict__ W2p, const float* __restrict__ b2,
          const float* __restrict__ W3, const float* __restrict__ b3,
          float* __restrict__ out) {
  constexpr int LDA = 390, LDH = 132;   // Kin=384
  __shared__ float As[16 * LDA];
  __shared__ float H1[16 * LDH];
  __shared__ float H2[16 * LDH];
  __shared__ int g0s[16], g1s[16];
  const int tid = threadIdx.x;
  const int r0 = blockIdx.x * 16;
  if (tid < 16) {
    const int j = r0 + tid;
    const int a0 = lei[j];
    const int a1 = lei[ELIG + j];
    g0s[tid] = (a0 / NLIG) * NG + (a0 % NLIG);
    g1s[tid] = (a1 / NLIG) * NG + (a1 % NLIG);
  }
  __syncthreads();
  for (int idx = tid; idx < 16 * LDA; idx += 256) {
    const int row = idx / LDA, col = idx % LDA;
    const int j = r0 + row;
    float v;
    if (col < 128)
      v = e[(size_t)(EKNN + j) * EDIM + col] + e[(size_t)(EKNN + HH + j) * EDIM + col];
    else if (col < 384)
      v = h[(size_t)g0s[row] * NDIM + (col - 128)] + h[(size_t)g1s[row] * NDIM + (col - 128)];
    else
      v = 0.0f;
    As[idx] = v;
  }
  __syncthreads();
  const int w = tid >> 5;
  v8f c0 = {};
  c0 = gemm_one(As, LDA, W1p, 128, 384, w * 16, c0);
  frag_store_act(H1, LDH, w * 16, c0, b1, 2);
  __syncthreads();
  v8f d0 = {};
  d0 = gemm_one(H1, LDH, W2p, 128, 128, w * 16, d0);
  frag_store_act(H2, LDH, w * 16, d0, b2, 2);
  __syncthreads();
  if (tid < 80) {
    const int r = tid / 5, c = tid % 5;
    float s = b3[c];
    for (int k = 0; k < 128; k++) s += H2[r * LDH + k] * W3[k * 5 + c];
    out[(size_t)(r0 + r) * 5 + c] = s;
  }
}

// ---------------------------------------------------------------------------
extern "C" void kernel_launch(void* const* d_in, const int* in_sizes, int n_in,
                              void* d_out, int out_size, void* d_ws, size_t ws_size,
                              hipStream_t stream) {
  const float* protein_node = (const float*)d_in[0];
  const float* protein_pos  = (const float*)d_in[1];
  const float* ligand_node  = (const float*)d_in[2];
  const float* ligand_pos   = (const float*)d_in[3];
  const int*   lig_edge_idx = (const int*)d_in[6];
  const int*   lig_edge_b   = (const int*)d_in[7];
  const int*   t            = (const int*)d_in[8];
  const float* W_pn = (const float*)d_in[9];
  const float* W_ln = (const float*)d_in[10];
  const float* W_le = (const float*)d_in[11];
  const float* W_pe = (const float*)d_in[12];
  const float* Wm1 = (const float*)d_in[13];
  const float* bm1 = (const float*)d_in[14];
  const float* Wm2 = (const float*)d_in[15];
  const float* bm2 = (const float*)d_in[16];
  const float* We1 = (const float*)d_in[17];
  const float* be1 = (const float*)d_in[18];
  const float* We2 = (const float*)d_in[19];
  const float* be2 = (const float*)d_in[20];
  const float* Wh1 = (const float*)d_in[21];
  const float* bh1 = (const float*)d_in[22];
  const float* Wh2 = (const float*)d_in[23];
  const float* bh2 = (const float*)d_in[24];
  const float* Wx  = (const float*)d_in[25];
  const float* dW1 = (const float*)d_in[26];
  const float* db1 = (const float*)d_in[27];
  const float* dW2 = (const float*)d_in[28];
  const float* db2 = (const float*)d_in[29];
  const float* dW3 = (const float*)d_in[30];
  const float* db3 = (const float*)d_in[31];
  float* out = (float*)d_out;

  char* p = (char*)d_ws;
  auto carve = [&](size_t bytes) {
    void* q = (void*)p;
    p += (bytes + 255) & ~(size_t)255;
    return q;
  };
  float* h     = (float*)carve((size_t)NNODE * NDIM * 4);
  float* posA  = (float*)carve((size_t)NNODE * 4 * 4);
  float* posB  = (float*)carve((size_t)NNODE * 4 * 4);
  float* e     = (float*)carve((size_t)ETOT * EDIM * 4);
  float* m     = (float*)carve((size_t)ETOT * 256 * 4);
  float* agg   = (float*)carve((size_t)NNODE * 256 * 4);
  float* dist2 = (float*)carve((size_t)ETOT * 4);
  float* et    = (float*)carve((size_t)ETOT * 4);
  float* vm    = (float*)carve((size_t)ETOT * 4);
  float* coef  = (float*)carve((size_t)ETOT * 4);
  int*   src   = (int*)carve((size_t)ETOT * 4);
  int*   dst   = (int*)carve((size_t)ETOT * 4);

  // packed weights: pairs = ceil4(Kin)/2
  const int PM1 = 322 * 256;   // Wm1: Kin 642 -> 644/2 = 322 pairs
  const int PM2 = 128 * 256;   // Wm2: Kin 256
  const int PE1 = 192 * 128;   // We1: Kin 384
  const int PE2 = 64 * 128;    // We2: Kin 128
  const int PH1 = 258 * 256;   // Wh1: Kin 513 -> 516/2 = 258 pairs
  const int PH2 = 128 * 256;   // Wh2: Kin 256
  const int PD1 = 192 * 128;   // dec W1: Kin 384
  const int PD2 = 64 * 128;    // dec W2: Kin 128
  float2* Wm1p = (float2*)carve((size_t)3 * PM1 * 8);
  float2* Wm2p = (float2*)carve((size_t)3 * PM2 * 8);
  float2* We1p = (float2*)carve((size_t)3 * PE1 * 8);
  float2* We2p = (float2*)carve((size_t)3 * PE2 * 8);
  float2* Wh1p = (float2*)carve((size_t)3 * PH1 * 8);
  float2* Wh2p = (float2*)carve((size_t)3 * PH2 * 8);
  float2* W1p  = (float2*)carve((size_t)PD1 * 8);
  float2* W2p  = (float2*)carve((size_t)PD2 * 8);

  auto pack = [&](const float* W, float2* Wp, int Kin, int N) {
    const int npairs = ((Kin + 3) & ~3) / 2;
    const int tot = npairs * N;
    k_pack_w<<<(tot + 255) / 256, 256, 0, stream>>>(W, Wp, Kin, N, npairs);
  };
  for (int l = 0; l < 3; l++) {
    pack(Wm1 + (size_t)l * 642 * 256, Wm1p + (size_t)l * PM1, 642, 256);
    pack(Wm2 + (size_t)l * 256 * 256, Wm2p + (size_t)l * PM2, 256, 256);
    pack(We1 + (size_t)l * 384 * 128, We1p + (size_t)l * PE1, 384, 128);
    pack(We2 + (size_t)l * 128 * 128, We2p + (size_t)l * PE2, 128, 128);
    pack(Wh1 + (size_t)l * 513 * 256, Wh1p + (size_t)l * PH1, 513, 256);
    pack(Wh2 + (size_t)l * 256 * 256, Wh2p + (size_t)l * PH2, 256, 256);
  }
  pack(dW1, W1p, 384, 128);
  pack(dW2, W2p, 128, 128);

  k_embed_nodes<<<NNODE, 256, 0, stream>>>(protein_node, protein_pos, ligand_node,
                                           ligand_pos, t, W_pn, W_ln, h, posA);
  k_knn<<<NNODE, NG, 0, stream>>>(posA, t, src, dst, vm, et);
  k_init_knn_edges<<<(EKNN * 128) / 256, 256, 0, stream>>>(vm, W_pe, e);
  k_embed_ligand_edges<<<ELIG, 128, 0, stream>>>(ligand_node, lig_edge_idx,
                                                 lig_edge_b, t, W_le, e, src, dst,
                                                 vm, et);

  float* pcur = posA;
  float* pnxt = posB;
  for (int l = 0; l < 3; l++) {
    k_edge_geom<<<(ETOT + 255) / 256, 256, 0, stream>>>(pcur, src, dst, dist2);
    k_msg<<<ETOT / 16, 256, 0, stream>>>(h, e, dist2, et, src, dst,
                                         Wm1p + (size_t)l * PM1, bm1 + l * 256,
                                         Wm2p + (size_t)l * PM2, bm2 + l * 256, m);
    k_eupd<<<ETOT / 16, 256, 0, stream>>>(e, m,
                                          We1p + (size_t)l * PE1, be1 + l * 128,
                                          We2p + (size_t)l * PE2, be2 + l * 128);
    k_agg<<<NNODE, 256, 0, stream>>>(m, vm, agg);
    k_hupd<<<NNODE / 16, 256, 0, stream>>>(h, agg, t,
                                           Wh1p + (size_t)l * PH1, bh1 + l * 256,
                                           Wh2p + (size_t)l * PH2, bh2 + l * 256);
    if (l < 2) {  // pos after the final layer is never consumed
      k_coef<<<ETOT / 8, 256, 0, stream>>>(m, Wx + l * 256, vm, coef);
      k_pos<<<(NNODE + 255) / 256, 256, 0, stream>>>(pcur, pnxt, dist2, coef, dst);
      float* tmp = pcur; pcur = pnxt; pnxt = tmp;
    }
  }

  k_decoder<<<HH / 16, 256, 0, stream>>>(h, e, lig_edge_idx, W1p, db1, W2p, db2,
                                         dW3, db3, out);
  (void)in_sizes; (void)n_in; (void)out_size; (void)ws_size;
}